// GraphSmoothingLoss_10376640987952
// MI455X (gfx1250) — compile-verified
//
#include <hip/hip_runtime.h>

// GraphSmoothingLoss for MI455X (gfx1250, wave32).
// 8-lanes-per-edge / 4-edges-per-wave gather layout:
//   - global_load_b128 row loads (4 contiguous 128B lines per instruction)
//   - ds_swizzle_b32 XOR-immediate reductions (no bpermute index setup)
//   - gfx1250 global_prefetch_b8 pipelining of next gathered rows
// Feature table (19.2 MB) is L2-resident (192 MB L2); kernel is L2-gather +
// instruction-issue bound, so minimize issue slots per edge.

#define D_FEAT 96

template<int PAT>
__device__ __forceinline__ float xor_swizzle_add(float v) {
    // ds_swizzle_b32 group-of-32 mode: offset[14:10]=xor, [9:5]=or, [4:0]=and
    return v + __int_as_float(
        __builtin_amdgcn_ds_swizzle(__float_as_int(v), PAT));
}

__global__ __launch_bounds__(256) void
graph_smoothing_loss_kernel(const float* __restrict__ feat,
                            const int*   __restrict__ eidx,
                            float*       __restrict__ out,
                            int nEdges, float invE)
{
    __shared__ float wave_sums[8];              // 8 waves per 256-thread block

    const int lane        = threadIdx.x & 31;   // wave32 on CDNA5
    const int g           = lane >> 3;          // edge slot within wave: 0..3
    const int sub         = lane & 7;           // lane within 8-lane group
    const int waveInBlock = threadIdx.x >> 5;
    const int wavesPerBlk = blockDim.x >> 5;
    const int gwave       = blockIdx.x * wavesPerBlk + waveInBlock;
    const int nWaves      = gridDim.x * wavesPerBlk;
    const int stride      = nWaves * 4;         // 4 edges per wave-iteration

    const int* __restrict__ srcIdx = eidx;          // edge_index[0]
    const int* __restrict__ dstIdx = eidx + nEdges; // edge_index[1]

    float acc = 0.0f;
    int base = gwave * 4;

    if (base < nEdges) {
        int  e     = base + g;
        bool valid = e < nEdges;
        int  ec    = valid ? e : 0;
        int   sCur = srcIdx[ec];
        int   dCur = dstIdx[ec];
        float mCur = valid ? 1.0f : 0.0f;
        float mNext = 0.0f;

        while (base < nEdges) {
            const float4* fs =
                (const float4*)(feat + (long)sCur * D_FEAT) + sub;
            const float4* fd =
                (const float4*)(feat + (long)dCur * D_FEAT) + sub;

            // Software pipeline: next quad's indices + row prefetch
            // (emits gfx1250 global_prefetch_b8) before chewing on this quad.
            const int baseN = base + stride;        // wave-uniform condition
            if (baseN < nEdges) {
                int  eN  = baseN + g;
                bool vN  = eN < nEdges;
                int  ecN = vN ? eN : 0;
                sCur  = srcIdx[ecN];
                dCur  = dstIdx[ecN];
                mNext = vN ? 1.0f : 0.0f;
                __builtin_prefetch(feat + (long)sCur * D_FEAT, 0, 0);
                __builtin_prefetch(feat + (long)dCur * D_FEAT, 0, 0);
            }

            // 3 x global_load_b128 per endpoint cover 4 edges' rows.
            float4 a0 = fs[0], a1 = fs[8], a2 = fs[16];
            float4 b0 = fd[0], b1 = fd[8], b2 = fd[16];

            float d, t;
            d = a0.x - b0.x; t = d * d;
            d = a0.y - b0.y; t = fmaf(d, d, t);
            d = a0.z - b0.z; t = fmaf(d, d, t);
            d = a0.w - b0.w; t = fmaf(d, d, t);
            d = a1.x - b1.x; t = fmaf(d, d, t);
            d = a1.y - b1.y; t = fmaf(d, d, t);
            d = a1.z - b1.z; t = fmaf(d, d, t);
            d = a1.w - b1.w; t = fmaf(d, d, t);
            d = a2.x - b2.x; t = fmaf(d, d, t);
            d = a2.y - b2.y; t = fmaf(d, d, t);
            d = a2.z - b2.z; t = fmaf(d, d, t);
            d = a2.w - b2.w; t = fmaf(d, d, t);

            // 3-step XOR butterfly within each 8-lane group; one ds_swizzle
            // instruction reduces all 4 edges at once.
            t = xor_swizzle_add<0x041F>(t);   // xor 1
            t = xor_swizzle_add<0x081F>(t);   // xor 2
            t = xor_swizzle_add<0x101F>(t);   // xor 4

            acc += sqrtf(t) * mCur;           // all 8 group lanes identical
            mCur = mNext;
            base = baseN;
        }
    }

    // Wave total: lanes within a group are duplicates, so only combine across
    // groups (lanes L, L^8, L^16, L^24) -> exact per-wave edge sum, no 8x.
    acc = xor_swizzle_add<0x201F>(acc);       // xor 8
    acc = xor_swizzle_add<0x401F>(acc);       // xor 16

    if (lane == 0)
        wave_sums[waveInBlock] = acc;
    __syncthreads();

    if (threadIdx.x == 0) {
        float blockSum = 0.0f;
        #pragma unroll
        for (int i = 0; i < 8; ++i)
            blockSum += wave_sums[i];
        atomicAdd(out, blockSum * invE);      // pre-scaled -> out = mean
    }
}

extern "C" void kernel_launch(void* const* d_in, const int* in_sizes, int n_in,
                              void* d_out, int out_size, void* d_ws, size_t ws_size,
                              hipStream_t stream)
{
    const float* feat = (const float*)d_in[0];   // (50000, 96) fp32
    const int*   eidx = (const int*)d_in[1];     // (2, E) int
    float*       out  = (float*)d_out;           // scalar loss

    const int nEdges = in_sizes[1] / 2;
    const float invE = 1.0f / (float)nEdges;

    // Deterministic per call: zero the accumulator, then accumulate.
    hipMemsetAsync(out, 0, sizeof(float), stream);

    const int threads = 256;                 // 8 waves/block (wave32)
    const int blocks  = 2048;                // 16384 waves x 4 edges in flight
    graph_smoothing_loss_kernel<<<blocks, threads, 0, stream>>>(
        feat, eidx, out, nEdges, invE);
}